// GATv2Denoiser_88055419502853
// MI455X (gfx1250) — compile-verified
//
#include <hip/hip_runtime.h>
#include <hip/hip_bf16.h>
#include <math.h>

// ---------------- problem constants (from reference setup_inputs) ------------
constexpr int B_   = 2;
constexpr int N_   = 20000;
constexpr int FIN  = 64;
constexpr int E_   = 320000;
constexpr int H_   = 4;
constexpr int C_   = 32;
constexpr int HC   = 128;          // H*C
constexpr int OUTC = 64;
constexpr int NT   = B_ * N_;      // 40000 nodes after block-diagonal merge
constexpr int ET   = B_ * E_ + NT; // 680000 edges incl. self loops

#define NEG_SLOPE 0.2f

typedef __attribute__((ext_vector_type(2))) float v2f;
typedef __attribute__((ext_vector_type(8))) float v8f;
typedef int v4i __attribute__((vector_size(16)));   // matches builtin param type

#if __has_builtin(__builtin_amdgcn_global_load_async_to_lds_b128) && \
    __has_builtin(__builtin_amdgcn_s_wait_asynccnt)
#define HAVE_ASYNC_LDS 1
typedef __attribute__((address_space(1))) v4i gv4i;   // global int4
typedef __attribute__((address_space(3))) v4i lv4i;   // LDS int4
#else
#define HAVE_ASYNC_LDS 0
#endif

// ---------------- edge list construction ------------------------------------
__global__ void build_edges_kernel(const long long* __restrict__ ei,
                                   int* __restrict__ src, int* __restrict__ dst) {
    int t = blockIdx.x * blockDim.x + threadIdx.x;
    if (t >= ET) return;
    if (t < B_ * E_) {
        int b = t / E_;
        int e = t - b * E_;
        long long s = ei[(long long)b * 2 * E_ + e]      + (long long)b * N_;
        long long d = ei[(long long)b * 2 * E_ + E_ + e] + (long long)b * N_;
        src[t] = (int)s;
        dst[t] = (int)d;
    } else {
        int n = t - B_ * E_;      // self loop
        src[t] = n;
        dst[t] = n;
    }
}

// ---------------- fp32 WMMA GEMM with LDS-staged A --------------------------
// One block per 16-row strip of Y. NW = NCOLS/16 waves; wave w computes the
// 16x16 tile at columns [16w, 16w+16). The 16xK A-strip (contiguous in memory)
// is staged once into LDS — via GLOBAL_LOAD_ASYNC_TO_LDS_B128 when the
// toolchain exposes it (ASYNCcnt path), else plain b128 load + ds_store.
// Per ISA 7.12.2:
//   A 16x4 f32: lane&15 = row, lanes>=16 hold K-halves {2,3} else {0,1}
//   C/D: VGPR r -> row r (lanes 0-15) / row 8+r (lanes 16-31), col = lane&15
template <int K, int NCOLS>
__global__ void __launch_bounds__(32 * (NCOLS / 16))
gemm_lds_wmma(const float* __restrict__ X, const float* __restrict__ W,
              const float* __restrict__ bias, float* __restrict__ Y) {
    constexpr int NW   = NCOLS / 16;
    constexpr int NTHR = 32 * NW;
    constexpr int TOT  = 16 * K;              // floats in the A strip

    __shared__ __align__(16) float As[TOT];

    const int tid  = threadIdx.x;
    const int lane = tid & 31;
    const int wv   = tid >> 5;
    const int mt   = blockIdx.x;

    const float* Xt = X + (size_t)mt * 16 * K;   // contiguous 16xK strip

    // ---- stage A strip into LDS ----
#if HAVE_ASYNC_LDS
#pragma unroll
    for (int i = tid * 4; i < TOT; i += NTHR * 4) {
        __builtin_amdgcn_global_load_async_to_lds_b128(
            (gv4i*)(Xt + i), (lv4i*)(As + i), 0, 0);
    }
    __builtin_amdgcn_s_wait_asynccnt(0);
#else
#pragma unroll
    for (int i = tid * 4; i < TOT; i += NTHR * 4) {
        *(float4*)(As + i) = *(const float4*)(Xt + i);
    }
#endif
    __syncthreads();

    // ---- WMMA K-loop ----
    const int rowA  = lane & 15;
    const int col   = wv * 16 + (lane & 15);
    const int khalf = (lane >> 4) * 2;        // 0 or 2

    v8f acc = {0.f, 0.f, 0.f, 0.f, 0.f, 0.f, 0.f, 0.f};

#pragma unroll 4
    for (int k0 = 0; k0 < K; k0 += 4) {
        v2f a, b;
        const float* ap = As + rowA * K + k0 + khalf;   // ds_load_b64
        a.x = ap[0];
        a.y = ap[1];
        const float* wp = W + (size_t)(k0 + khalf) * NCOLS + col;
        b.x = wp[0];
        b.y = wp[NCOLS];
        acc = __builtin_amdgcn_wmma_f32_16x16x4_f32(
                  /*neg_a=*/false, a, /*neg_b=*/false, b,
                  /*c_mod=*/(short)0, acc, /*reuse_a=*/false, /*reuse_b=*/false);
    }

    const float bv = bias ? bias[col] : 0.f;
    const int rbase = mt * 16 + (lane >> 4) * 8;
#pragma unroll
    for (int r = 0; r < 8; ++r) {
        Y[(size_t)(rbase + r) * NCOLS + col] = acc[r] + bv;
    }
}

// ---------------- per-layer state init --------------------------------------
__global__ void init_layer_kernel(float* __restrict__ m, float* __restrict__ denom,
                                  float* __restrict__ agg) {
    int i = blockIdx.x * blockDim.x + threadIdx.x;
    if (i < NT * H_) {
        m[i]     = -INFINITY;
        denom[i] = 0.f;
    }
    if (i < NT * HC) agg[i] = 0.f;
}

// ---------------- atomic float max via CAS ----------------------------------
__device__ __forceinline__ void atomicMaxF(float* addr, float val) {
    int* ia = (int*)addr;
    int old = __float_as_int(*addr);
    while (__int_as_float(old) < val) {
        int assumed = old;
        old = atomicCAS(ia, assumed, __float_as_int(val));
        if (old == assumed) break;
    }
}

__device__ __forceinline__ float leaky(float x) {
    return x > 0.f ? x : NEG_SLOPE * x;
}

// ---------------- edge scores: e = att . leakyrelu(xl[src]+xr[dst]) ---------
__global__ void edge_scores_kernel(const int* __restrict__ src, const int* __restrict__ dst,
                                   const float* __restrict__ xl, const float* __restrict__ xr,
                                   const float* __restrict__ att,
                                   float* __restrict__ e, float* __restrict__ m) {
    int t = blockIdx.x * blockDim.x + threadIdx.x;
    if (t >= ET * H_) return;
    const int h  = t & (H_ - 1);
    const int ed = t >> 2;           // H_ == 4
    const int s  = src[ed];
    const int d  = dst[ed];
    const float4* pl = (const float4*)(xl + (size_t)s * HC + h * C_);
    const float4* pr = (const float4*)(xr + (size_t)d * HC + h * C_);
    const float4* pa = (const float4*)(att + h * C_);
    float acc = 0.f;
#pragma unroll
    for (int c = 0; c < C_ / 4; ++c) {
        float4 vl = pl[c], vr = pr[c], va = pa[c];
        acc += va.x * leaky(vl.x + vr.x);
        acc += va.y * leaky(vl.y + vr.y);
        acc += va.z * leaky(vl.z + vr.z);
        acc += va.w * leaky(vl.w + vr.w);
    }
    e[t] = acc;
    atomicMaxF(&m[(size_t)d * H_ + h], acc);
}

// ---------------- a = exp(e - m[dst]); denom[dst] += a (in place over e) ----
__global__ void edge_exp_kernel(const int* __restrict__ dst, const float* __restrict__ m,
                                float* __restrict__ e, float* __restrict__ denom) {
    int t = blockIdx.x * blockDim.x + threadIdx.x;
    if (t >= ET * H_) return;
    const int h  = t & (H_ - 1);
    const int ed = t >> 2;
    const int d  = dst[ed];
    float a = __expf(e[t] - m[(size_t)d * H_ + h]);
    e[t] = a;
    atomicAdd(&denom[(size_t)d * H_ + h], a);
}

// ---------------- agg[dst] += (a/denom[dst]) * xl[src] ----------------------
__global__ void edge_aggregate_kernel(const int* __restrict__ src, const int* __restrict__ dst,
                                      const float* __restrict__ a, const float* __restrict__ denom,
                                      const float* __restrict__ xl, float* __restrict__ agg) {
    int t = blockIdx.x * blockDim.x + threadIdx.x;
    if (t >= ET * H_) return;
    const int h  = t & (H_ - 1);
    const int ed = t >> 2;
    const int s  = src[ed];
    const int d  = dst[ed];
    const float alpha = a[t] / denom[(size_t)d * H_ + h];
    const float* pl = xl + (size_t)s * HC + h * C_;
    float*       po = agg + (size_t)d * HC + h * C_;
    __builtin_prefetch(pl + C_, 0, 1);   // gfx1250 global_prefetch_b8
#pragma unroll
    for (int c = 0; c < C_; ++c) {
        atomicAdd(po + c, alpha * pl[c]);   // global_atomic_add_f32
    }
}

// ---------------- h = elu(agg + bias), in place -----------------------------
__global__ void finalize_elu_kernel(float* __restrict__ agg, const float* __restrict__ bias) {
    int i = blockIdx.x * blockDim.x + threadIdx.x;
    if (i >= NT * HC) return;
    float v = agg[i] + bias[i & (HC - 1)];
    agg[i] = v > 0.f ? v : expm1f(v);
}

// ---------------- orchestration ---------------------------------------------
extern "C" void kernel_launch(void* const* d_in, const int* in_sizes, int n_in,
                              void* d_out, int out_size, void* d_ws, size_t ws_size,
                              hipStream_t stream) {
    const float*     x    = (const float*)d_in[0];
    const long long* ei   = (const long long*)d_in[1];   // int64 per reference
    const float*     Wl1  = (const float*)d_in[2];
    const float*     Wr1  = (const float*)d_in[3];
    const float*     att1 = (const float*)d_in[4];
    const float*     b1   = (const float*)d_in[5];
    const float*     Wl2  = (const float*)d_in[6];
    const float*     Wr2  = (const float*)d_in[7];
    const float*     att2 = (const float*)d_in[8];
    const float*     b2   = (const float*)d_in[9];
    const float*     Wo   = (const float*)d_in[10];
    const float*     bo   = (const float*)d_in[11];
    float*           out  = (float*)d_out;

    // workspace carve-up
    char*  ws  = (char*)d_ws;
    size_t off = 0;
    auto carve = [&](size_t bytes) -> void* {
        void* p = ws + off;
        off = (off + bytes + 255) & ~(size_t)255;
        return p;
    };
    int*   src   = (int*)  carve(sizeof(int)   * ET);
    int*   dst   = (int*)  carve(sizeof(int)   * ET);
    float* xl    = (float*)carve(sizeof(float) * (size_t)NT * HC);
    float* xr    = (float*)carve(sizeof(float) * (size_t)NT * HC);
    float* e     = (float*)carve(sizeof(float) * (size_t)ET * H_);
    float* m     = (float*)carve(sizeof(float) * (size_t)NT * H_);
    float* denom = (float*)carve(sizeof(float) * (size_t)NT * H_);
    float* agg1  = (float*)carve(sizeof(float) * (size_t)NT * HC);
    float* agg2  = (float*)carve(sizeof(float) * (size_t)NT * HC);

    const int TPB = 256;
    dim3 blk(TPB);
    dim3 gEdges((ET + TPB - 1) / TPB);
    dim3 gEH((ET * H_ + TPB - 1) / TPB);
    dim3 gNode((NT * HC + TPB - 1) / TPB);
    dim3 gStrip(NT / 16);       // 2500 blocks, one per 16-row strip

    // 0) edges
    build_edges_kernel<<<gEdges, blk, 0, stream>>>(ei, src, dst);

    // 1) layer 1
    gemm_lds_wmma<FIN, HC><<<gStrip, dim3(32 * (HC / 16)), 0, stream>>>(x, Wl1, nullptr, xl);
    gemm_lds_wmma<FIN, HC><<<gStrip, dim3(32 * (HC / 16)), 0, stream>>>(x, Wr1, nullptr, xr);
    init_layer_kernel<<<gNode, blk, 0, stream>>>(m, denom, agg1);
    edge_scores_kernel   <<<gEH, blk, 0, stream>>>(src, dst, xl, xr, att1, e, m);
    edge_exp_kernel      <<<gEH, blk, 0, stream>>>(dst, m, e, denom);
    edge_aggregate_kernel<<<gEH, blk, 0, stream>>>(src, dst, e, denom, xl, agg1);
    finalize_elu_kernel  <<<gNode, blk, 0, stream>>>(agg1, b1);

    // 2) layer 2 (input = agg1)
    gemm_lds_wmma<HC, HC><<<gStrip, dim3(32 * (HC / 16)), 0, stream>>>(agg1, Wl2, nullptr, xl);
    gemm_lds_wmma<HC, HC><<<gStrip, dim3(32 * (HC / 16)), 0, stream>>>(agg1, Wr2, nullptr, xr);
    init_layer_kernel<<<gNode, blk, 0, stream>>>(m, denom, agg2);
    edge_scores_kernel   <<<gEH, blk, 0, stream>>>(src, dst, xl, xr, att2, e, m);
    edge_exp_kernel      <<<gEH, blk, 0, stream>>>(dst, m, e, denom);
    edge_aggregate_kernel<<<gEH, blk, 0, stream>>>(src, dst, e, denom, xl, agg2);
    finalize_elu_kernel  <<<gNode, blk, 0, stream>>>(agg2, b2);

    // 3) output projection: out = h2 @ Wo + bo
    gemm_lds_wmma<HC, OUTC><<<gStrip, dim3(32 * (OUTC / 16)), 0, stream>>>(agg2, Wo, bo, out);
}